// EdgeNet_17583596110112
// MI455X (gfx1250) — compile-verified
//
#include <hip/hip_runtime.h>
#include <hip/hip_bf16.h>

typedef __attribute__((ext_vector_type(2))) float v2f;
typedef __attribute__((ext_vector_type(8))) float v8f;

#define N_NODES 2048

// ---------------------------------------------------------------------------
// K1: per-node: norm, u = x^2/norm, and fill Hc[:, 0:36] (=h padded) + Hc[:,71]=0
// Hc layout: [N][72] = [ x(32) | cent(3) | 0 | neigh1(35 written later) | 0 ]
// ---------------------------------------------------------------------------
__global__ void node_prep(const float* __restrict__ x, const float* __restrict__ cent,
                          float* __restrict__ norms, float* __restrict__ u,
                          float* __restrict__ Hc) {
    int node = blockIdx.x * (blockDim.x >> 5) + (threadIdx.x >> 5);
    int lane = threadIdx.x & 31;
    if (node >= N_NODES) return;
    float v = x[node * 32 + lane];
    float sq = v * v;
    float s = sq;
    for (int d = 16; d > 0; d >>= 1) s += __shfl_xor(s, d, 32);
    float nrm = sqrtf(s);
    if (lane == 0) norms[node] = nrm;
    u[node * 32 + lane] = sq / nrm;
    Hc[node * 72 + lane] = v;
    if (lane < 3) Hc[node * 72 + 32 + lane] = cent[node * 3 + lane];
    if (lane == 3) Hc[node * 72 + 35] = 0.0f;
    if (lane == 4) Hc[node * 72 + 71] = 0.0f;
}

// ---------------------------------------------------------------------------
// Two-level exclusive prefix-scan over nodes, features in lanes.
// scan_local: 64 blocks x F threads, 32 nodes per block (exclusive within block)
// scan_blocks: single block of F threads scans the 64 block sums.
// ---------------------------------------------------------------------------
__global__ void scan_local(const float* __restrict__ src, float* __restrict__ pre,
                           float* __restrict__ bsum, int F) {
    int b = blockIdx.x;
    int k = threadIdx.x;
    int base = b * 32;
    float carry = 0.0f;
    for (int jj = 0; jj < 32; ++jj) {
        float v = src[(size_t)(base + jj) * F + k];
        pre[(size_t)(base + jj) * F + k] = carry;
        carry += v;
    }
    bsum[b * F + k] = carry;
}

__global__ void scan_blocks(const float* __restrict__ bsum, float* __restrict__ bpre,
                            int F, int nb) {
    int k = threadIdx.x;
    float carry = 0.0f;
    for (int b = 0; b < nb; ++b) {
        float v = bsum[b * F + k];
        bpre[b * F + k] = carry;
        carry += v;
    }
}

// ---------------------------------------------------------------------------
// K3: centroid part of neigh1: Hc[j][68+f] = (1/j) * sum_{i<j} c[i,f]*|c[i,f]-c[j,f]|
// one block (256 thr) per j
// ---------------------------------------------------------------------------
__global__ void cdiff_kernel(const float* __restrict__ cent, float* __restrict__ Hc) {
    int j = blockIdx.x;
    float cj0 = cent[j * 3 + 0], cj1 = cent[j * 3 + 1], cj2 = cent[j * 3 + 2];
    float a0 = 0.f, a1 = 0.f, a2 = 0.f;
    for (int i = threadIdx.x; i < j; i += 256) {
        float c0 = cent[i * 3 + 0], c1 = cent[i * 3 + 1], c2 = cent[i * 3 + 2];
        a0 += c0 * fabsf(c0 - cj0);
        a1 += c1 * fabsf(c1 - cj1);
        a2 += c2 * fabsf(c2 - cj2);
    }
    __shared__ float red[3 * 256];
    red[threadIdx.x] = a0; red[256 + threadIdx.x] = a1; red[512 + threadIdx.x] = a2;
    __syncthreads();
    for (int s = 128; s > 0; s >>= 1) {
        if ((int)threadIdx.x < s) {
            red[threadIdx.x]       += red[threadIdx.x + s];
            red[256 + threadIdx.x] += red[256 + threadIdx.x + s];
            red[512 + threadIdx.x] += red[512 + threadIdx.x + s];
        }
        __syncthreads();
    }
    if (threadIdx.x == 0) {
        float inv = (j > 0) ? 1.0f / (float)j : 0.0f;
        Hc[j * 72 + 68] = red[0]   * inv;
        Hc[j * 72 + 69] = red[256] * inv;
        Hc[j * 72 + 70] = red[512] * inv;
    }
}

// ---------------------------------------------------------------------------
// K4: x-part of neigh1 via prefix sums: Hc[j][36+k] = x[j,k]*P[j,k]/(norm[j]*j)
// ---------------------------------------------------------------------------
__global__ void neighx_kernel(const float* __restrict__ x, const float* __restrict__ norms,
                              const float* __restrict__ P1, const float* __restrict__ bp1,
                              float* __restrict__ Hc) {
    int idx = blockIdx.x * blockDim.x + threadIdx.x;
    if (idx >= N_NODES * 32) return;
    int j = idx >> 5, k = idx & 31;
    float val = 0.0f;
    if (j > 0) {
        float p = P1[idx] + bp1[(j >> 5) * 32 + k];
        val = x[idx] * p / (norms[j] * (float)j);
    }
    Hc[j * 72 + 36 + k] = val;
}

// ---------------------------------------------------------------------------
// Weight packers
// ---------------------------------------------------------------------------
__global__ void build_Wc1(const float* __restrict__ Ws, const float* __restrict__ Wn,
                          float* __restrict__ Wc) {  // [72][64]
    int idx = blockIdx.x * blockDim.x + threadIdx.x;
    if (idx >= 72 * 64) return;
    int r = idx / 64, c = idx % 64;
    float v = 0.0f;
    if (r < 35) v = Ws[r * 64 + c];
    else if (r >= 36 && r < 71) v = Wn[(r - 36) * 64 + c];
    Wc[idx] = v;
}

__global__ void build_Wc2(const float* __restrict__ Ws, const float* __restrict__ Wn,
                          float* __restrict__ Wc) {  // [128][32]
    int idx = blockIdx.x * blockDim.x + threadIdx.x;
    if (idx >= 128 * 32) return;
    int r = idx / 32, c = idx % 32;
    Wc[idx] = (r < 64) ? Ws[r * 32 + c] : Wn[(r - 64) * 32 + c];
}

// Wm1c[32][64]: cols 0..31 = Wm1_top (rows 0..31 of Wm1[64,32]); cols 32..63 = Wm1_bot.
// Also biasAB[64] (bm1 on first 32 cols) and wvec[32] = Wm2[:,1]-Wm2[:,0].
__global__ void build_head(const float* __restrict__ Wm1, const float* __restrict__ bm1,
                           const float* __restrict__ Wm2, float* __restrict__ Wm1c,
                           float* __restrict__ biasAB, float* __restrict__ wvec) {
    int idx = blockIdx.x * blockDim.x + threadIdx.x;
    if (idx < 32 * 64) {
        int r = idx / 64, c = idx % 64;
        Wm1c[idx] = (c < 32) ? Wm1[r * 32 + c] : Wm1[(32 + r) * 32 + (c - 32)];
    }
    if (idx < 64) biasAB[idx] = (idx < 32) ? bm1[idx] : 0.0f;
    if (idx < 32) wvec[idx] = Wm2[idx * 2 + 1] - Wm2[idx * 2 + 0];
}

// ---------------------------------------------------------------------------
// Generic fp32 WMMA GEMM: C[M,N] = A[M,K] @ B[K,N] + bias[N]
// One wave per 16x16 tile; V_WMMA_F32_16X16X4_F32, 4-wide K steps.
// A frag (16x4): lane<16 -> row=lane, {K0,K1}; lane>=16 -> row=lane-16, {K2,K3}
// B frag (4x16): lane<16 -> col=lane, {K0,K1}; lane>=16 -> col=lane-16, {K2,K3}
// C frag: vgpr r -> row = r + 8*(lane/16), col = lane%16
// ---------------------------------------------------------------------------
__global__ void wmma_gemm_f32(const float* __restrict__ A, const float* __restrict__ B,
                              const float* __restrict__ bias, float* __restrict__ C,
                              int M, int N, int K) {
    int wave = (blockIdx.x * blockDim.x + threadIdx.x) >> 5;
    int lane = threadIdx.x & 31;
    int ntilesN = N >> 4;
    int mt = wave / ntilesN;
    int nt = wave % ntilesN;
    if (mt >= (M >> 4)) return;
    int rc = lane & 15;     // row (for A) / col (for B/C) within tile
    int half = lane >> 4;
    v8f acc = {};
    const float* arow = A + (size_t)(mt * 16 + rc) * K;
    const float* bcol = B + (nt * 16 + rc);
    for (int kk = 0; kk < K; kk += 4) {
        int ka = kk + 2 * half;
        v2f a = *(const v2f*)(arow + ka);          // K even, ka even -> 8B aligned
        v2f b;
        b.x = bcol[(size_t)ka * N];
        b.y = bcol[(size_t)(ka + 1) * N];
        acc = __builtin_amdgcn_wmma_f32_16x16x4_f32(false, a, false, b,
                                                    (short)0, acc, false, false);
    }
    int col = nt * 16 + rc;
    float bv = bias[col];
    #pragma unroll
    for (int r = 0; r < 8; ++r) {
        int row = mt * 16 + r + 8 * half;
        C[(size_t)row * N + col] = acc[r] + bv;
    }
}

// ---------------------------------------------------------------------------
// K7: Hc2[N][128] = [ h1 | neigh2 ], neigh2 from h1 prefix sums / j
// ---------------------------------------------------------------------------
__global__ void build_Hc2(const float* __restrict__ h1, const float* __restrict__ P2,
                          const float* __restrict__ bp2, float* __restrict__ Hc2) {
    int idx = blockIdx.x * blockDim.x + threadIdx.x;
    if (idx >= N_NODES * 128) return;
    int j = idx >> 7, k = idx & 127;
    float v;
    if (k < 64) v = h1[j * 64 + k];
    else {
        int kk = k - 64;
        v = (j > 0) ? (P2[j * 64 + kk] + bp2[(j >> 5) * 64 + kk]) / (float)j : 0.0f;
    }
    Hc2[idx] = v;
}

// ---------------------------------------------------------------------------
// K11: edge head. Tile 16(i) x 64(j). AB[i][0:32]=A-part(+bm1), AB[i][32:64]=B-part.
// out[e] = softmax over 2 classes == (sigmoid(-d), sigmoid(d)), d = relu(A+B)·w + cb
// LDS rows padded to 36 floats: 16B-aligned rows; a wave32 b128 read at stride-36
// DWORDs hits every bank exactly twice = the 2-cycle minimum (no extra conflicts).
// As reads are wave-uniform (broadcast). Output stored as one float2 per edge.
// ---------------------------------------------------------------------------
__global__ void edge_head(const float* __restrict__ AB, const float* __restrict__ wvec,
                          const float* __restrict__ bm2, float* __restrict__ out) {
    const int NN = N_NODES;
    int i0 = blockIdx.y * 16;
    int j0 = blockIdx.x * 64;
    if (i0 >= j0 + 64) return;           // whole tile below diagonal
    __shared__ float As[16][36];
    __shared__ float Bs[64][36];
    __shared__ float ws[32];
    int tid = threadIdx.x;               // 256
    for (int t = tid; t < 16 * 32; t += 256) {
        int li = t >> 5, k = t & 31;
        As[li][k] = AB[(i0 + li) * 64 + k];
    }
    for (int t = tid; t < 64 * 32; t += 256) {
        int lj = t >> 5, k = t & 31;
        Bs[lj][k] = AB[(j0 + lj) * 64 + 32 + k];
    }
    if (tid < 32) ws[tid] = wvec[tid];
    __syncthreads();
    float cb = bm2[1] - bm2[0];
    #pragma unroll
    for (int it = 0; it < 4; ++it) {
        int t2 = it * 256 + tid;
        int li = t2 >> 6, lj = t2 & 63;  // li uniform across each wave
        int i = i0 + li, j = j0 + lj;
        if (i < j) {
            float acc = 0.0f;
            #pragma unroll
            for (int k4 = 0; k4 < 8; ++k4) {
                float4 av = *(const float4*)(&As[li][k4 * 4]);   // broadcast b128
                float4 bv = *(const float4*)(&Bs[lj][k4 * 4]);   // conflict-min b128
                float4 wv = *(const float4*)(&ws[k4 * 4]);       // broadcast b128
                acc = fmaf(fmaxf(av.x + bv.x, 0.0f), wv.x, acc);
                acc = fmaf(fmaxf(av.y + bv.y, 0.0f), wv.y, acc);
                acc = fmaf(fmaxf(av.z + bv.z, 0.0f), wv.z, acc);
                acc = fmaf(fmaxf(av.w + bv.w, 0.0f), wv.w, acc);
            }
            float d = acc + cb;
            float e = __expf(d);
            float o0 = 1.0f / (1.0f + e);          // softmax class 0
            int eidx = i * (NN - 1) - (i * (i - 1)) / 2 + (j - i - 1);
            float2 o2; o2.x = o0; o2.y = 1.0f - o0;
            *(float2*)(out + (size_t)eidx * 2) = o2;   // 8B-aligned coalesced store
        }
    }
}

// ---------------------------------------------------------------------------
extern "C" void kernel_launch(void* const* d_in, const int* in_sizes, int n_in,
                              void* d_out, int out_size, void* d_ws, size_t ws_size,
                              hipStream_t stream) {
    const float* x     = (const float*)d_in[0];   // [2048,32]
    const float* cent  = (const float*)d_in[1];   // [2048,3]
    const float* Ws1   = (const float*)d_in[2];   // [35,64]
    const float* Wn1   = (const float*)d_in[3];   // [35,64]
    const float* b1    = (const float*)d_in[4];   // [64]
    const float* Ws2   = (const float*)d_in[5];   // [64,32]
    const float* Wn2   = (const float*)d_in[6];   // [64,32]
    const float* b2    = (const float*)d_in[7];   // [32]
    const float* Wm1   = (const float*)d_in[8];   // [64,32]
    const float* bm1   = (const float*)d_in[9];   // [32]
    const float* Wm2   = (const float*)d_in[10];  // [32,2]
    const float* bm2   = (const float*)d_in[11];  // [2]
    float* out = (float*)d_out;

    const int N = N_NODES;
    float* W = (float*)d_ws;
    size_t o = 0;
    auto alloc = [&](size_t n) { float* p = W + o; o += n; return p; };
    float* norms = alloc(N);            // 2048
    float* u     = alloc((size_t)N * 32);
    float* P1    = alloc((size_t)N * 32);
    float* bs1   = alloc(64 * 32);
    float* bp1   = alloc(64 * 32);
    float* Hc    = alloc((size_t)N * 72);
    float* Wc1   = alloc(72 * 64);
    float* h1    = alloc((size_t)N * 64);
    float* P2    = alloc((size_t)N * 64);
    float* bs2   = alloc(64 * 64);
    float* bp2   = alloc(64 * 64);
    float* Hc2   = alloc((size_t)N * 128);
    float* Wc2   = alloc(128 * 32);
    float* h2    = alloc((size_t)N * 32);
    float* Wm1c  = alloc(32 * 64);
    float* biasAB= alloc(64);
    float* wvec  = alloc(32);
    float* AB    = alloc((size_t)N * 64);
    (void)ws_size; (void)in_sizes; (void)n_in; (void)out_size;

    // node features
    node_prep<<<N / 8, 256, 0, stream>>>(x, cent, norms, u, Hc);
    // prefix sums of u (32 feats)
    scan_local<<<64, 32, 0, stream>>>(u, P1, bs1, 32);
    scan_blocks<<<1, 32, 0, stream>>>(bs1, bp1, 32, 64);
    // centroid aggregation (3 feats, O(N^2/2))
    cdiff_kernel<<<N, 256, 0, stream>>>(cent, Hc);
    // x-part of neigh1
    neighx_kernel<<<(N * 32) / 256, 256, 0, stream>>>(x, norms, P1, bp1, Hc);
    // layer 1 GEMM: [2048x72] @ [72x64] + b1
    build_Wc1<<<(72 * 64 + 255) / 256, 256, 0, stream>>>(Ws1, Wn1, Wc1);
    wmma_gemm_f32<<<(128 * 4) / 4, 128, 0, stream>>>(Hc, Wc1, b1, h1, N, 64, 72);
    // prefix sums of h1 (64 feats) -> neigh2, build Hc2
    scan_local<<<64, 64, 0, stream>>>(h1, P2, bs2, 64);
    scan_blocks<<<1, 64, 0, stream>>>(bs2, bp2, 64, 64);
    build_Hc2<<<(N * 128) / 256, 256, 0, stream>>>(h1, P2, bp2, Hc2);
    // layer 2 GEMM: [2048x128] @ [128x32] + b2
    build_Wc2<<<(128 * 32 + 255) / 256, 256, 0, stream>>>(Ws2, Wn2, Wc2);
    wmma_gemm_f32<<<(128 * 2) / 4, 128, 0, stream>>>(Hc2, Wc2, b2, h2, N, 32, 128);
    // head precompute: AB = h2 @ Wm1c + biasAB  ([2048x32]@[32x64])
    build_head<<<(32 * 64 + 255) / 256, 256, 0, stream>>>(Wm1, bm1, Wm2, Wm1c, biasAB, wvec);
    wmma_gemm_f32<<<(128 * 4) / 4, 128, 0, stream>>>(h2, Wm1c, biasAB, AB, N, 64, 32);
    // per-edge head: 2.1M edges
    dim3 egrid(N / 64, N / 16);
    edge_head<<<egrid, 256, 0, stream>>>(AB, wvec, bm2, out);
}